// CommDiffMambaBlock_35081292873778
// MI455X (gfx1250) — compile-verified
//
#include <hip/hip_runtime.h>
#include <hip/hip_bf16.h>
#include <math.h>

// ---------------- problem constants (from reference setup_inputs) -----------
#define B_   4
#define S_   2
#define C_   96
#define H_   32
#define W_   32
#define STR  3          // stream = S + 1
#define G_   12         // 4 * stream
#define N_   16         // D_STATE
#define R_   6          // LOW_RANK
#define P_   38         // N + N + R
#define PPAD 48         // P padded to 3 WMMA tiles
#define L_   2048       // S * H * W
#define OUTSZ (B_ * S_ * H_ * W_ * C_)   // 786432 per output tensor

typedef float v2f __attribute__((ext_vector_type(2)));
typedef float v8f __attribute__((ext_vector_type(8)));
typedef unsigned int u32x4 __attribute__((ext_vector_type(4)));
typedef int i32x4 __attribute__((ext_vector_type(4)));
typedef int i32x8 __attribute__((ext_vector_type(8)));

// ---------------------------------------------------------------------------
// TDM: issue a 2D tensor_load_to_lds. D# built per CDNA5 ISA §8.3/§8.4:
//  g0: count=1 | lds_addr | global_addr(57b) | type=2
//  g1: data_size=4B | tensor_dim0/1 | tile_dim0/1 | tensor_dim0_stride
// Groups 2/3 zero (2D tensor). Flat LDS addresses truncate to the 32-bit
// LDS byte offset.
// This toolchain's builtin takes 6 args (g0, g1, g2, g3, g4, cpol).
// ---------------------------------------------------------------------------
__device__ __forceinline__ void tdm_load_2d(void* lds_dst, const void* gsrc,
                                            unsigned tensor_d0, unsigned tensor_d1,
                                            unsigned tile_d0, unsigned tile_d1,
                                            unsigned long d0_stride_elems) {
  unsigned long ga = (unsigned long)(uintptr_t)gsrc;
  unsigned lds_off = (unsigned)(uintptr_t)lds_dst;
  u32x4 g0;
  g0[0] = 1u;                                          // count=1 (valid), user mode
  g0[1] = lds_off;                                     // lds_addr [63:32]
  g0[2] = (unsigned)(ga & 0xFFFFFFFFu);                // global_addr low
  g0[3] = (unsigned)((ga >> 32) & 0x01FFFFFFu) | (2u << 30);  // addr hi | type=2
  i32x8 g1;
  g1[0] = (int)(2u << 16);                             // data_size = 2 (4 bytes)
  g1[1] = (int)((tensor_d0 & 0xFFFFu) << 16);          // tensor_dim0 [15:0]
  g1[2] = (int)((tensor_d0 >> 16) & 0xFFFFu) |
          (int)((tensor_d1 & 0xFFFFu) << 16);          // dim0 hi | dim1 lo
  g1[3] = (int)((tensor_d1 >> 16) & 0xFFFFu) |
          (int)((tile_d0 & 0xFFFFu) << 16);            // dim1 hi | tile_dim0
  g1[4] = (int)(tile_d1 & 0xFFFFu);                    // tile_dim1, tile_dim2=0
  g1[5] = (int)(d0_stride_elems & 0xFFFFFFFFu);        // dim0_stride low32
  g1[6] = (int)((d0_stride_elems >> 32) & 0xFFFFu);    // dim0_stride hi16
  g1[7] = 0;
  i32x4 z4 = {0, 0, 0, 0};
  i32x8 z8 = {0, 0, 0, 0, 0, 0, 0, 0};
  __builtin_amdgcn_tensor_load_to_lds(g0, g1, z4, z4, z8, 0);
}

// ---------------------------------------------------------------------------
// Kernel 1: build xs (B, 4, stream, C, L) from x.
// ---------------------------------------------------------------------------
__global__ void k_build_xs(const float* __restrict__ x, float* __restrict__ XS) {
  long idx = (long)blockIdx.x * blockDim.x + threadIdx.x;
  const long total = (long)B_ * G_ * C_ * L_;
  if (idx >= total) return;
  int l = (int)(idx % L_); long t = idx / L_;
  int c = (int)(t % C_);   t /= C_;
  int g = (int)(t % G_);   int b = (int)(t / G_);
  int k = g / STR, s = g % STR;
  int kk = k, ll = l;
  if (k >= 2) { kk = k - 2; ll = L_ - 1 - l; }
  int sg = ll % S_; int q = ll / S_;
  int h, w;
  if (kk == 0) { h = q / W_; w = q % W_; } else { w = q / H_; h = q % H_; }
  const long xcs = (long)C_ * H_ * W_;
  const float* xb = x + (long)b * S_ * xcs + (long)c * H_ * W_ + h * W_ + w;
  float v;
  if (s == STR - 1) {
    v = xb[(long)sg * xcs];
  } else {
    float xv = xb[(long)s * xcs];
    if (sg == S_ - 1)      v = xv;
    else if (sg == s)      v = xv - xb[(long)(S_ - 1) * xcs];
    else                   v = xv - xb[(long)sg * xcs];
  }
  XS[idx] = v;
}

// ---------------------------------------------------------------------------
// Kernel 2: x_dbl[b,g,p,l] = sum_c W[g,p,c] * xs[b,g,c,l]
// Block = 96 threads (3 waves), one (bg, ntile) per block.
// TDM stages both operands into LDS; A-panel zero-padded to 48 rows so the
// inner loop has NO masks/branches. Wave w owns m-tile w; 24 K=4 WMMA steps.
// ---------------------------------------------------------------------------
__global__ void k_proj_gemm(const float* __restrict__ XS,
                            const float* __restrict__ Wp,
                            float* __restrict__ XDBL) {
  __shared__ float apanel[PPAD * C_];                  // 18 KB, rows P_..47 zero
  __shared__ float bpanel[C_ * 16];                    // 6 KB
  const int tid   = threadIdx.x;                       // 0..95
  const int lane  = tid & 31;
  const int wid   = tid >> 5;                          // 0..2 -> m-tile
  const int nbase = blockIdx.x * 16;
  const int bg    = blockIdx.y;
  const int g     = bg % G_;
  const float* Wm = Wp + (long)g * P_ * C_;
  const float* Xm = XS + (long)bg * C_ * L_;
  float*       Om = XDBL + (long)bg * P_ * L_;

  // zero the pad rows of the A panel (disjoint from TDM target rows 0..P_-1)
  for (int i = tid; i < (PPAD - P_) * C_; i += 96) apanel[P_ * C_ + i] = 0.f;
  if (wid == 0) {                                      // one wave issues both DMAs
    tdm_load_2d(bpanel, Xm + nbase, L_, C_, 16, C_, (unsigned long)L_);
    tdm_load_2d(apanel, Wm, C_, P_, C_, P_, (unsigned long)C_);
    __builtin_amdgcn_s_wait_tensorcnt(0);
  }
  __syncthreads();

  const int half = lane >> 4, lm = lane & 15;
  const int mrow = wid * 16 + lm;
  v8f acc = {0.f, 0.f, 0.f, 0.f, 0.f, 0.f, 0.f, 0.f};
  for (int kk = 0; kk < C_; kk += 4) {
    int ka = kk + 2 * half;                            // this lane-half's K pair
    v2f a, bf;
    a.x  = apanel[mrow * C_ + ka];
    a.y  = apanel[mrow * C_ + ka + 1];
    bf.x = bpanel[ka * 16 + lm];
    bf.y = bpanel[(ka + 1) * 16 + lm];
    acc = __builtin_amdgcn_wmma_f32_16x16x4_f32(false, a, false, bf,
                                                (short)0, acc, false, false);
  }
#pragma unroll
  for (int r = 0; r < 8; ++r) {
    int m = wid * 16 + r + 8 * half;
    if (m < P_) Om[(long)m * L_ + nbase + lm] = acc[r];
  }
}

// ---------------------------------------------------------------------------
// Kernel 3: dts[b,g,c,l] = sum_r Wd[g,c,r] * x_dbl[b,g,r,l]   (K=6 pad to 8)
// Branch-free masking: clamp row index + multiply A by 0/1.
// ---------------------------------------------------------------------------
__global__ void k_dt_gemm(const float* __restrict__ XDBL,
                          const float* __restrict__ Wd,
                          float* __restrict__ DTS) {
  const int lane  = threadIdx.x;
  const int nbase = blockIdx.x * 16;
  const int mbase = blockIdx.y * 16;
  const int bg    = blockIdx.z;
  const int g     = bg % G_;
  const float* Wm  = Wd + (long)g * C_ * R_;
  const float* Din = XDBL + (long)bg * P_ * L_;        // rows 0..R-1
  float*       Om  = DTS + (long)bg * C_ * L_;
  const int half = lane >> 4, lm = lane & 15;
  const int mrow = mbase + lm;
  v8f acc = {0.f, 0.f, 0.f, 0.f, 0.f, 0.f, 0.f, 0.f};
#pragma unroll
  for (int kk = 0; kk < 8; kk += 4) {
    int ka = kk + 2 * half;
    float m0 = (ka     < R_) ? 1.f : 0.f;
    float m1 = (ka + 1 < R_) ? 1.f : 0.f;
    int   r0 = (ka     < R_) ? ka     : 0;             // clamp -> valid address
    int   r1 = (ka + 1 < R_) ? ka + 1 : 0;
    v2f a, bf;
    a.x  = m0 * Wm[mrow * R_ + r0];                    // zero pad via VALU mask
    a.y  = m1 * Wm[mrow * R_ + r1];
    bf.x = Din[(long)r0 * L_ + nbase + lm];
    bf.y = Din[(long)r1 * L_ + nbase + lm];
    acc = __builtin_amdgcn_wmma_f32_16x16x4_f32(false, a, false, bf,
                                                (short)0, acc, false, false);
  }
#pragma unroll
  for (int r = 0; r < 8; ++r) {
    int m = mbase + r + 8 * half;                      // C_ multiple of 16
    Om[(long)m * L_ + nbase + lm] = acc[r];
  }
}

// ---------------------------------------------------------------------------
// Kernel 4: selective scan, state-parallel. One lane per (channel, n):
// 73728 lanes = 2304 wave32s. Per step: 1 B-load + 1 C-load + 1 exp per lane;
// y = sum_n h*c via 4-step shfl_xor butterfly inside each 16-lane group.
// ys overwrites xs in place (all group lanes read u before lane 0 stores).
// ---------------------------------------------------------------------------
__global__ void k_scan(float* XSYS,                        // in: u, out: ys
                       const float* __restrict__ XDBL,
                       const float* __restrict__ DTS,
                       const float* __restrict__ dt_bias,  // (G*C,)
                       const float* __restrict__ A_logs,   // (G*C, N)
                       const float* __restrict__ Ds) {     // (G*C,)
  const int n  = threadIdx.x & 15;                         // state index
  const int ch = blockIdx.x * (blockDim.x >> 4) + (threadIdx.x >> 4);
  const int c  = ch % C_;
  const int bg = ch / C_;
  const int g  = bg % G_;
  float*       uy   = XSYS + (long)ch * L_;
  const float* d_in = DTS  + (long)ch * L_;
  const float* Brow = XDBL + ((long)bg * P_ + R_ + n) * L_;
  const float* Crow = XDBL + ((long)bg * P_ + R_ + N_ + n) * L_;
  const float  bias = dt_bias[g * C_ + c];
  const float  dsv  = Ds[g * C_ + c];
  const float  An   = -__expf(A_logs[(long)(g * C_ + c) * N_ + n]);
  const bool   writer = (n == 0);
  float h = 0.f;
  for (int l = 0; l < L_; ++l) {
    __builtin_prefetch(Brow + l + 64, 0, 0);               // global_prefetch_b8
    __builtin_prefetch(Crow + l + 64, 0, 0);
    float draw  = d_in[l] + bias;                          // broadcast load
    float delta = (draw > 20.f) ? draw : log1pf(__expf(draw));  // softplus
    float u     = uy[l];                                   // broadcast load
    h = h * __expf(delta * An) + (delta * u) * Brow[l];
    float y = h * Crow[l];
    y += __shfl_xor(y, 1, 32);                             // reduce over n
    y += __shfl_xor(y, 2, 32);
    y += __shfl_xor(y, 4, 32);
    y += __shfl_xor(y, 8, 32);
    if (writer) uy[l] = y + dsv * u;                       // ys in place
  }
}

// ---------------------------------------------------------------------------
// Kernel 5: direction-combine + inverse scans + 3 fused layernorms.
// ---------------------------------------------------------------------------
__device__ __forceinline__ float ys_dir(const float* YS, int b, int k, int s,
                                        int c, int l) {
  long b1 = (((long)b * G_ + (k * STR + s)) * C_ + c) * (long)L_;
  long b2 = (((long)b * G_ + ((k + 2) * STR + s)) * C_ + c) * (long)L_;
  return YS[b1 + l] + YS[b2 + (L_ - 1 - l)];
}

__device__ __forceinline__ float yd_f(const float* YS, int b, int st, int sg,
                                      int c, int h, int w) {
  int l0 = (h * W_ + w) * S_ + sg;
  int l1 = (w * H_ + h) * S_ + sg;
  return ys_dir(YS, b, 0, st, c, l0) + ys_dir(YS, b, 1, st, c, l1);
}

__global__ void k_out(const float* __restrict__ YS,
                      const float* __restrict__ nw,
                      const float* __restrict__ nb,
                      float* __restrict__ out) {
  const int lane = threadIdx.x;                        // 0..31 (wave32)
  const int pos  = blockIdx.x * blockDim.y + threadIdx.y;
  int w = pos % W_; int t = pos / W_;
  int h = t % H_;   t /= H_;
  int st = t % S_;  int b = t / S_;

  float vals[3][3];
#pragma unroll
  for (int j = 0; j < 3; ++j) {
    int c = lane + 32 * j;
    float ydv[S_];
    float sumyd = 0.f;
#pragma unroll
    for (int sg = 0; sg < S_; ++sg) {
      ydv[sg] = yd_f(YS, b, st, sg, c, h, w);
      sumyd += ydv[sg];
    }
    vals[0][j] = ydv[S_ - 1];                                        // y
    vals[1][j] = ys_dir(YS, b, 0, STR - 1, c, (h * W_ + w) * S_ + st)
               + ys_dir(YS, b, 1, STR - 1, c, (w * H_ + h) * S_ + st); // y_com
    vals[2][j] = sumyd - ((st < S_ - 1) ? ydv[S_ - 1] : ydv[st]);    // y_dif
  }

  const long pbase = (((long)(b * S_ + st) * H_ + h) * W_ + w) * C_;
#pragma unroll
  for (int o = 0; o < 3; ++o) {
    float s = 0.f, sq = 0.f;
#pragma unroll
    for (int j = 0; j < 3; ++j) { s += vals[o][j]; sq += vals[o][j] * vals[o][j]; }
#pragma unroll
    for (int m = 16; m > 0; m >>= 1) {                 // wave32 butterfly
      s  += __shfl_xor(s, m, 32);
      sq += __shfl_xor(sq, m, 32);
    }
    float mean = s * (1.0f / C_);
    float var  = sq * (1.0f / C_) - mean * mean;
    float inv  = rsqrtf(var + 1e-5f);
#pragma unroll
    for (int j = 0; j < 3; ++j) {
      int c = lane + 32 * j;
      out[(long)o * OUTSZ + pbase + c] =
          (vals[o][j] - mean) * inv * nw[c] + nb[c];
    }
  }
}

// ---------------------------------------------------------------------------
extern "C" void kernel_launch(void* const* d_in, const int* in_sizes, int n_in,
                              void* d_out, int out_size, void* d_ws, size_t ws_size,
                              hipStream_t stream) {
  const float* x     = (const float*)d_in[0];
  const float* xpw   = (const float*)d_in[1];
  const float* dtw   = (const float*)d_in[2];
  const float* dtb   = (const float*)d_in[3];
  const float* alogs = (const float*)d_in[4];
  const float* Dsp   = (const float*)d_in[5];
  const float* nw    = (const float*)d_in[6];
  const float* nb    = (const float*)d_in[7];
  float* out = (float*)d_out;

  // workspace layout (floats): XS (reused as YS) | XDBL | DTS  (~90 MB < L2)
  float* XS   = (float*)d_ws;
  float* XDBL = XS + (size_t)B_ * G_ * C_ * L_;        // 9,437,184 floats
  float* DTS  = XDBL + (size_t)B_ * G_ * P_ * L_;      // +3,735,552 floats

  {
    long total = (long)B_ * G_ * C_ * L_;
    k_build_xs<<<(int)((total + 255) / 256), 256, 0, stream>>>(x, XS);
  }
  {
    dim3 grid(L_ / 16, B_ * G_);                       // 128 x 48 blocks
    k_proj_gemm<<<grid, 96, 0, stream>>>(XS, xpw, XDBL);
  }
  {
    dim3 grid(L_ / 16, C_ / 16, B_ * G_);              // 128 x 6 x 48 tiles
    k_dt_gemm<<<grid, 32, 0, stream>>>(XDBL, dtw, DTS);
  }
  {
    int channels = B_ * G_ * C_;                       // 4608
    dim3 blk(128);                                     // 8 channels / block
    k_scan<<<channels / 8, blk, 0, stream>>>(XS, XDBL, DTS, dtb, alogs, Dsp);
  }
  {
    dim3 blk(32, 8);                                   // 8 pixels per block
    int positions = B_ * S_ * H_ * W_;                 // 8192
    k_out<<<positions / 8, blk, 0, stream>>>(XS, nw, nb, out);
  }
}